// AnyNet_73658689126858
// MI455X (gfx1250) — compile-verified
//
#include <hip/hip_runtime.h>

// Problem shape (fixed by setup_inputs)
#define B_   8
#define C_   8
#define S_   16
#define H_   128
#define W_   416
#define HW_  (H_ * W_)                 // 53248
#define THREADS 256
#define PPT  2                          // pixels per thread
#define PIX_PER_WG (THREADS * PPT)      // 512  -> 104 tiles per batch exactly
#define LDS_PAD 4                       // 16B-aligned pad so clamped tap index -1 stays in-bounds
#define LDS_FLOATS (HW_ + 2 * LDS_PAD)

// ---- CDNA5 data-mover paths (guarded; toolchains differ) ----
#if defined(__gfx1250__) && \
    __has_builtin(__builtin_amdgcn_tensor_load_to_lds) && \
    __has_builtin(__builtin_amdgcn_s_wait_tensorcnt)
#define HAS_TDM 1
#else
#define HAS_TDM 0
#endif

#if defined(__gfx1250__) && \
    __has_builtin(__builtin_amdgcn_global_load_async_to_lds_b128) && \
    __has_builtin(__builtin_amdgcn_s_wait_asynccnt)
#define HAS_ASYNC_LDS 1
#else
#define HAS_ASYNC_LDS 0
#endif

typedef unsigned u32x4 __attribute__((ext_vector_type(4)));
typedef int      i32x8 __attribute__((ext_vector_type(8)));
typedef int      i32x4 __attribute__((ext_vector_type(4)));
typedef int      v4i_  __attribute__((ext_vector_type(4)));
typedef __attribute__((address_space(1))) v4i_*  gv4_ptr;
typedef __attribute__((address_space(3))) v4i_*  lv4_ptr;
typedef __attribute__((address_space(3))) float* lds_fptr;

__device__ __forceinline__ unsigned pack_h2(float a, float b) {
  unsigned short ua = __builtin_bit_cast(unsigned short, (_Float16)a);
  unsigned short ub = __builtin_bit_cast(unsigned short, (_Float16)b);
  return (unsigned)ua | ((unsigned)ub << 16);
}
__device__ __forceinline__ float h2_lo(unsigned u) {
  return (float)__builtin_bit_cast(_Float16, (unsigned short)(u & 0xffffu));
}
__device__ __forceinline__ float h2_hi(unsigned u) {
  return (float)__builtin_bit_cast(_Float16, (unsigned short)(u >> 16));
}

// Issue the copy of one 213KB feat_r channel slice into LDS (no wait here).
__device__ __forceinline__ void stage_issue(const float* __restrict__ src,
                                            float* __restrict__ dst, int tid) {
#if HAS_TDM
  // One TDM descriptor moves the whole slice: 1 row of HW_ 4-byte elements.
  // D# per CDNA5 ISA 8.3/8.4: group0 = {flags, lds_addr, global_addr, type=2},
  // group1 = {data_size=4B, tensor_dim0/1, tile_dim0/1, strides}; groups 2/3
  // zero (2D tensor: tile_dim2/3 = 0 -> unused). Issued by wave 0 only
  // (EXEC is ignored by TDM; one instruction = one DMA).
  if (tid < 32) {
    const unsigned long long ga = (unsigned long long)(const void*)src;
    const unsigned lds_byte = (unsigned)(unsigned long long)(lds_fptr)dst;
    const unsigned td0 = (unsigned)HW_;
    u32x4 g0 = { 1u,                                    // count=1, user mode
                 lds_byte,                              // lds_addr (bytes)
                 (unsigned)(ga & 0xffffffffu),          // global_addr[31:0]
                 (unsigned)((ga >> 32) & 0x01ffffffu)   // global_addr[56:32]
                   | (2u << 30) };                      // type=2 ("image")
    i32x8 g1 = { (int)(2u << 16),                       // data_size=2 -> 4 bytes
                 (int)((td0 & 0xffffu) << 16),          // tensor_dim0[15:0] @bit48
                 (int)((td0 >> 16) | (1u << 16)),       // tensor_dim0[31:16] | tensor_dim1=1
                 (int)((td0 & 0xffffu) << 16),          // tensor_dim1[31:16]=0 | tile_dim0
                 1,                                     // tile_dim1=1, tile_dim2=0
                 (int)td0,                              // tensor_dim0_stride[31:0]
                 (int)((td0 & 0xffffu) << 16),          // stride0[47:32]=0 | stride1[15:0]
                 (int)(td0 >> 16) };                    // stride1[47:16]=0
    i32x4 gz4 = { 0, 0, 0, 0 };
    i32x8 gz8 = { 0, 0, 0, 0, 0, 0, 0, 0 };
    __builtin_amdgcn_tensor_load_to_lds(g0, g1, gz4, gz4, gz8, 0);
  }
#elif HAS_ASYNC_LDS
  // 16B per lane per instruction; 416 wave-instructions (52 per wave) cover HW_ exactly.
#pragma unroll 4
  for (int ofs = tid * 4; ofs < HW_; ofs += THREADS * 4) {
    __builtin_amdgcn_global_load_async_to_lds_b128(
        (gv4_ptr)(src + ofs), (lv4_ptr)(dst + ofs), 0, 0);
  }
#else
  for (int ofs = tid * 4; ofs < HW_; ofs += THREADS * 4) {
    const float4 v = *(const float4*)(src + ofs);
    *(float4*)(dst + ofs) = v;
  }
#endif
}

__device__ __forceinline__ void stage_wait(int tid) {
#if HAS_TDM
  if (tid < 32) __builtin_amdgcn_s_wait_tensorcnt(0);
#elif HAS_ASYNC_LDS
  __builtin_amdgcn_s_wait_asynccnt(0);  // each wave drains its own asynccnt
#endif
  (void)tid;
}

__global__ __launch_bounds__(THREADS, 1)
void anynet_cost_volume(const float* __restrict__ feat_l,
                        const float* __restrict__ feat_r,
                        const float* __restrict__ lut,
                        const unsigned char* __restrict__ vmask,
                        float* __restrict__ out) {
  extern __shared__ float sm[];  // LDS_FLOATS floats: one feat_r channel slice
  const int tid  = threadIdx.x;
  const int b    = blockIdx.y;
  const int pix0 = blockIdx.x * PIX_PER_WG;
  float* dst = sm + LDS_PAD;     // 16B-aligned staging destination

  // Kick off the c=0 slice copy immediately; it overlaps the VALU-heavy
  // parameter-packing prologue below.
  stage_issue(feat_r + (size_t)(b * C_ + 0) * HW_, dst, tid);

  // Per-(pixel,s) precomputed tap state, kept in VGPRs across the c loop.
  unsigned pbase[PPT][S_];  // low16 = LDS index of v00 row, high16 = of v10 row
  unsigned wab[PPT][S_];    // f16x2 (w00, w01) with zero-padding folded in
  unsigned wcd[PPT][S_];    // f16x2 (w10, w11)
  float    acc[PPT][S_];

#pragma unroll
  for (int p = 0; p < PPT; ++p) {
    const int pix = pix0 + p * THREADS + tid;
#pragma unroll
    for (int s = 0; s < S_; ++s) {
      const size_t li = ((size_t)(b * S_ + s) * HW_ + pix) * 3;
      const float gx = lut[li + 0];
      const float gy = lut[li + 1];
      // unnormalize, align_corners=False
      const float ix = ((gx + 1.0f) * (float)W_ - 1.0f) * 0.5f;
      const float iy = ((gy + 1.0f) * (float)H_ - 1.0f) * 0.5f;
      const float fx = floorf(ix);
      const float fy = floorf(iy);
      const float tx = ix - fx;
      const float ty = iy - fy;
      const int x0 = (int)fx;
      const int y0 = (int)fy;
      // zero-padding validity folded into weights
      const float vx0 = (x0 >= 0 && x0 < W_) ? 1.0f : 0.0f;
      const float vx1 = (x0 + 1 >= 0 && x0 + 1 < W_) ? 1.0f : 0.0f;
      const float vy0 = (y0 >= 0 && y0 < H_) ? 1.0f : 0.0f;
      const float vy1 = (y0 + 1 >= 0 && y0 + 1 < H_) ? 1.0f : 0.0f;
      // clamp so reads stay inside the padded LDS slice; invalid taps weigh 0
      const int xc = min(max(x0, -1), W_ - 1);
      const int r0 = min(max(y0, 0), H_ - 1);
      const int r1 = min(max(y0 + 1, 0), H_ - 1);
      const unsigned b0 = (unsigned)(r0 * W_ + xc + LDS_PAD);  // < 65536
      const unsigned b1 = (unsigned)(r1 * W_ + xc + LDS_PAD);
      pbase[p][s] = b0 | (b1 << 16);
      const float w00 = (1.0f - tx) * (1.0f - ty) * vx0 * vy0;
      const float w01 = tx * (1.0f - ty) * vx1 * vy0;
      const float w10 = (1.0f - tx) * ty * vx0 * vy1;
      const float w11 = tx * ty * vx1 * vy1;
      wab[p][s] = pack_h2(w00, w01);
      wcd[p][s] = pack_h2(w10, w11);
      acc[p][s] = 0.0f;
    }
  }

  for (int c = 0; c < C_; ++c) {
    stage_wait(tid);     // issuing wave(s) drain their copy counter
    __syncthreads();     // slice visible to all 8 waves

#pragma unroll
    for (int p = 0; p < PPT; ++p) {
      const int pix = pix0 + p * THREADS + tid;
      const float fl = feat_l[(size_t)(b * C_ + c) * HW_ + pix];
      const unsigned char* mp = vmask + (size_t)(b * C_ + c) * S_ * HW_ + pix;
#pragma unroll
      for (int s = 0; s < S_; ++s) {
        const unsigned pb = pbase[p][s];
        const int i0 = (int)(pb & 0xffffu);
        const int i1 = (int)(pb >> 16);
        // adjacent pairs -> ds_load_2addr_b32 (offsets n, n+1)
        const float v00 = sm[i0];
        const float v01 = sm[i0 + 1];
        const float v10 = sm[i1];
        const float v11 = sm[i1 + 1];
        const unsigned wa = wab[p][s];
        const unsigned wc = wcd[p][s];
        const float warped = v00 * h2_lo(wa) + v01 * h2_hi(wa) +
                             v10 * h2_lo(wc) + v11 * h2_hi(wc);
        const float d = fabsf(warped - fl);
        const bool valid = mp[(size_t)s * HW_] != 0;
        acc[p][s] += valid ? d : __builtin_inff();
      }
    }

    __syncthreads();     // all readers done before the slice is overwritten
    if (c + 1 < C_) {
      stage_issue(feat_r + (size_t)(b * C_ + (c + 1)) * HW_, dst, tid);
    }
  }

#pragma unroll
  for (int p = 0; p < PPT; ++p) {
    const int pix = pix0 + p * THREADS + tid;
#pragma unroll
    for (int s = 0; s < S_; ++s) {
      out[(size_t)(b * S_ + s) * HW_ + pix] = acc[p][s];
    }
  }
}

extern "C" void kernel_launch(void* const* d_in, const int* in_sizes, int n_in,
                              void* d_out, int out_size, void* d_ws, size_t ws_size,
                              hipStream_t stream) {
  (void)in_sizes; (void)n_in; (void)out_size; (void)d_ws; (void)ws_size;
  const float* feat_l        = (const float*)d_in[0];
  const float* feat_r        = (const float*)d_in[1];
  const float* lut           = (const float*)d_in[2];
  const unsigned char* vmask = (const unsigned char*)d_in[3];  // jnp bool -> 1 byte
  // d_in[4] = no_steps (S_=16, compile-time constant)
  float* out = (float*)d_out;

  const size_t smem = (size_t)LDS_FLOATS * sizeof(float);  // ~213 KB (fits 320 KB WGP LDS)
  (void)hipFuncSetAttribute((const void*)anynet_cost_volume,
                            hipFuncAttributeMaxDynamicSharedMemorySize, (int)smem);

  dim3 grid(HW_ / PIX_PER_WG, B_);  // (104, 8)
  anynet_cost_volume<<<grid, THREADS, smem, stream>>>(feat_l, feat_r, lut, vmask, out);
}